// ScaledDotProductAttention_13211319403250
// MI455X (gfx1250) — compile-verified
//
#include <hip/hip_runtime.h>

typedef __attribute__((ext_vector_type(2))) float    v2f;
typedef __attribute__((ext_vector_type(4))) float    v4f;
typedef __attribute__((ext_vector_type(8))) float    v8f;
typedef __attribute__((ext_vector_type(4))) unsigned v4u;
typedef __attribute__((address_space(3))) float lds_f;

constexpr int BATCH = 2;
constexpr int HEADS = 16;
constexpr int SEQ   = 2048;
constexpr int DIM   = 64;
constexpr int TOPK  = 32;
constexpr int SROW  = SEQ + 4;        // 2052 floats; 2052%64==4 -> conflict-free A gathers
constexpr int KROW  = DIM + 4;        // 68 floats = 272 B (16B multiple -> aligned b128 LDS writes)
constexpr int KTILE = 16 * KROW;      // 1088 floats per K tile buffer
constexpr float INV_SCALE = 0.125f;   // 1/8.0

__device__ __forceinline__ unsigned fkey(float f) {
    // monotonic float -> uint key (larger key <=> larger float)
    unsigned u = __float_as_uint(f);
    return u ^ ((u >> 31) ? 0xFFFFFFFFu : 0x80000000u);
}
__device__ __forceinline__ float fdec(unsigned key) {
    // inverse of fkey
    unsigned u = (key & 0x80000000u) ? (key ^ 0x80000000u) : ~key;
    return __uint_as_float(u);
}
__device__ __forceinline__ unsigned umaxu(unsigned a, unsigned b) { return a > b ? a : b; }
__device__ __forceinline__ unsigned uminu(unsigned a, unsigned b) { return a < b ? a : b; }

// ---------------------------------------------------------------------------
// Fully fused top-k attention. One block = 16 query rows of one (b,h); 8 waves.
//   P1: scores = (Q@K^T)/8 via fp32 WMMA; K tiles double-buffered into LDS
//       with global_load_async_to_lds_b128 (ASYNCcnt); A-fragments hoisted.
//   P2: exact top-32 via ballot-count radix select (key domain, early exit)
//       + masked softmax; NT-stream attn/mask.
//   P3: context = attn@V via fp32 WMMA, attn read from LDS.
// ---------------------------------------------------------------------------
__global__ __launch_bounds__(256) void fused_topk_attn(
    const float* __restrict__ q, const float* __restrict__ k,
    const float* __restrict__ v, float* __restrict__ ctx,
    float* __restrict__ attn, float* __restrict__ mask)
{
    __shared__ __align__(16) float sc[16 * SROW];          // 131,328 B score slab
    __shared__ __align__(16) float kbuf[8 * 2 * KTILE];    //  69,632 B K double-buffers
                                                           // (reused for P3 partials)

    const int bh      = blockIdx.x >> 7;          // / (SEQ/16)
    const int rowTile = blockIdx.x & 127;
    const int wave    = threadIdx.x >> 5;
    const int lane    = threadIdx.x & 31;
    const int m       = lane & 15;                // M (A) / N (B) index
    const int ksel    = (lane >> 4) * 2;          // K-pair select (ISA A/B layout)
    const int rowSel  = lane >> 4;

    const float* Q  = q + (size_t)bh * SEQ * DIM + (size_t)(rowTile * 16) * DIM;
    const float* Kp = k + (size_t)bh * SEQ * DIM;
    const float* Vp = v + (size_t)bh * SEQ * DIM;

    // 32-bit LDS byte address of kbuf (addrspace(3) ptrtoint)
    const unsigned kbufAddr = (unsigned)(unsigned long long)(lds_f*)&kbuf[0];
    const unsigned myBuf    = kbufAddr + (unsigned)(wave * 2 * KTILE) * 4u;

    auto issueTile = [&](int ct, int dbl) {
        const float* g = Kp + (size_t)(ct * 16) * DIM + lane * 4;
        const unsigned base = myBuf + (unsigned)(dbl * KTILE) * 4u;
#pragma unroll
        for (int i = 0; i < 8; ++i) {
            const unsigned lds = base + (unsigned)(((2 * i + rowSel) * KROW + m * 4) * 4);
            const float* gp = g + i * 128;
            asm volatile("global_load_async_to_lds_b128 %0, %1, off"
                         :: "v"(lds), "v"(gp) : "memory");
        }
    };

    // ---------------- Phase 1: scores into LDS (double-buffered K) ----------
    // Hoist the 16 A-fragments (Q is block-invariant) into registers once.
    v2f aF[16];
#pragma unroll
    for (int dd = 0; dd < 16; ++dd)
        aF[dd] = *(const v2f*)&Q[m * DIM + dd * 4 + ksel];

    issueTile(wave, 0);
    for (int j = 0; j < 16; ++j) {                 // 128 tiles / 8 waves
        const int cur = j & 1;
        if (j < 15) {
            issueTile(wave + 8 * (j + 1), cur ^ 1);
            asm volatile("s_wait_asynccnt 0x8" ::: "memory");  // tile j landed
        } else {
            asm volatile("s_wait_asynccnt 0x0" ::: "memory");
        }
        const float* kt = &kbuf[wave * 2 * KTILE + cur * KTILE];
        v8f c = {0.f, 0.f, 0.f, 0.f, 0.f, 0.f, 0.f, 0.f};
#pragma unroll
        for (int d0 = 0; d0 < DIM; d0 += 4) {
            v2f b = *(const v2f*)&kt[m * KROW + d0 + ksel];   // ds_load, banks (4n+d)%64
            c = __builtin_amdgcn_wmma_f32_16x16x4_f32(false, aF[d0 / 4], false, b,
                                                      (short)0, c, false, false);
        }
        const int ct   = wave + 8 * j;
        const int rowb = (lane < 16) ? 0 : 8;
#pragma unroll
        for (int r = 0; r < 8; ++r)
            sc[(rowb + r) * SROW + ct * 16 + m] = c[r] * INV_SCALE;
    }
    __syncthreads();

    // ------- Phase 2: per-row exact top-32 + masked softmax (2 rows/wave) ----
    for (int r = wave * 2; r < wave * 2 + 2; ++r) {
        const v4f* srow4 = (const v4f*)&sc[r * SROW];
        v4u k4[16];                                // sortable keys, 64 per lane
#pragma unroll
        for (int i = 0; i < 16; ++i) {
            const v4f sv = srow4[i * 32 + lane];
            k4[i].x = fkey(sv.x); k4[i].y = fkey(sv.y);
            k4[i].z = fkey(sv.z); k4[i].w = fkey(sv.w);
        }

        // row max in key domain (monotonic bijection)
        unsigned mk = 0u;
#pragma unroll
        for (int i = 0; i < 16; ++i)
            mk = umaxu(umaxu(k4[i].x, k4[i].y), umaxu(umaxu(k4[i].z, k4[i].w), mk));
        for (int off = 16; off; off >>= 1) mk = umaxu(mk, __shfl_xor(mk, off, 32));
        const float mx = fdec(mk);

        // radix select: ballot-popcount counting, early exit when the
        // candidate subgroup size equals `want` (threshold = subgroup min)
        unsigned prefix = 0u, hiFix = 0u;
        int want = TOPK;
        bool found = false;
        for (int bit = 31; bit >= 0 && !found; --bit) {
            const unsigned test = prefix | (1u << bit);
            const unsigned hi   = ~((1u << bit) - 1u);
            int cnt = 0;
#pragma unroll
            for (int i = 0; i < 16; ++i) {
                cnt += __popcll(__ballot((k4[i].x & hi) == test));
                cnt += __popcll(__ballot((k4[i].y & hi) == test));
                cnt += __popcll(__ballot((k4[i].z & hi) == test));
                cnt += __popcll(__ballot((k4[i].w & hi) == test));
            }
            if (cnt >= want) {
                prefix = test; hiFix = hi;
                if (cnt == want) found = true;
            } else {
                want -= cnt;
            }
        }
        unsigned tk;
        if (found) {
            unsigned mn = 0xFFFFFFFFu;
#pragma unroll
            for (int i = 0; i < 16; ++i) {
                if ((k4[i].x & hiFix) == prefix) mn = uminu(mn, k4[i].x);
                if ((k4[i].y & hiFix) == prefix) mn = uminu(mn, k4[i].y);
                if ((k4[i].z & hiFix) == prefix) mn = uminu(mn, k4[i].z);
                if ((k4[i].w & hiFix) == prefix) mn = uminu(mn, k4[i].w);
            }
            for (int off = 16; off; off >>= 1) mn = uminu(mn, __shfl_xor(mn, off, 32));
            tk = mn;
        } else {
            tk = prefix;          // all 32 bits resolved: prefix is the key
        }

        // sum of exp over kept entries (key-domain compare; ties kept)
        float sum = 0.f;
#pragma unroll
        for (int i = 0; i < 16; ++i) {
            sum += (k4[i].x >= tk) ? __expf(fdec(k4[i].x) - mx) : 0.f;
            sum += (k4[i].y >= tk) ? __expf(fdec(k4[i].y) - mx) : 0.f;
            sum += (k4[i].z >= tk) ? __expf(fdec(k4[i].z) - mx) : 0.f;
            sum += (k4[i].w >= tk) ? __expf(fdec(k4[i].w) - mx) : 0.f;
        }
        for (int off = 16; off; off >>= 1) sum += __shfl_xor(sum, off, 32);
        const float inv = 1.f / sum;

        const int grow = bh * SEQ + rowTile * 16 + r;
        v4f* arow = (v4f*)(attn + (size_t)grow * SEQ);
        v4f* mrow = (v4f*)(mask + (size_t)grow * SEQ);
        v4f* lrow = (v4f*)&sc[r * SROW];
#pragma unroll
        for (int i = 0; i < 16; ++i) {
            const v4u kv = k4[i];
            v4f av, mv;
            av.x = (kv.x >= tk) ? __expf(fdec(kv.x) - mx) * inv : 0.f;
            av.y = (kv.y >= tk) ? __expf(fdec(kv.y) - mx) * inv : 0.f;
            av.z = (kv.z >= tk) ? __expf(fdec(kv.z) - mx) * inv : 0.f;
            av.w = (kv.w >= tk) ? __expf(fdec(kv.w) - mx) * inv : 0.f;
            mv.x = (kv.x < tk) ? 1.f : 0.f;
            mv.y = (kv.y < tk) ? 1.f : 0.f;
            mv.z = (kv.z < tk) ? 1.f : 0.f;
            mv.w = (kv.w < tk) ? 1.f : 0.f;
            lrow[i * 32 + lane] = av;                               // attn back to LDS
            __builtin_nontemporal_store(av, &arow[i * 32 + lane]);  // streamed outputs
            __builtin_nontemporal_store(mv, &mrow[i * 32 + lane]);
        }
    }
    __syncthreads();

    // ---------------- Phase 3: context = attn @ V (fp32 WMMA) ----------------
    float* part = &kbuf[0];        // reuse K-buffer LDS for partials (post-barrier)
    const int t    = wave & 3;
    const int half = wave >> 2;
    v8f c = {0.f, 0.f, 0.f, 0.f, 0.f, 0.f, 0.f, 0.f};
    const int k0 = half * (SEQ / 2);
    const int k1 = k0 + (SEQ / 2);
    for (int kk = k0; kk < k1; kk += 4) {
        v2f a, b;
        a = *(const v2f*)&sc[m * SROW + kk + ksel];   // padded rows -> clean banks
        b.x = Vp[(size_t)(kk + ksel) * DIM + t * 16 + m];
        b.y = Vp[(size_t)(kk + ksel + 1) * DIM + t * 16 + m];
        c = __builtin_amdgcn_wmma_f32_16x16x4_f32(false, a, false, b,
                                                  (short)0, c, false, false);
    }
#pragma unroll
    for (int r = 0; r < 8; ++r) part[wave * 256 + r * 32 + lane] = c[r];
    __syncthreads();

    if (wave < 4) {
        float* out = ctx + (size_t)bh * SEQ * DIM + (size_t)(rowTile * 16) * DIM;
        const int rowb = (lane < 16) ? 0 : 8;
#pragma unroll
        for (int r = 0; r < 8; ++r) {
            const float val = part[wave * 256 + r * 32 + lane]
                            + part[(wave + 4) * 256 + r * 32 + lane];
            __builtin_nontemporal_store(val, &out[(size_t)(rowb + r) * DIM + t * 16 + m]);
        }
    }
}

// ---------------------------------------------------------------------------
extern "C" void kernel_launch(void* const* d_in, const int* in_sizes, int n_in,
                              void* d_out, int out_size, void* d_ws, size_t ws_size,
                              hipStream_t stream) {
    (void)in_sizes; (void)n_in; (void)out_size; (void)d_ws; (void)ws_size;
    const float* q = (const float*)d_in[0];
    const float* k = (const float*)d_in[1];
    const float* v = (const float*)d_in[2];

    float* ctx  = (float*)d_out;
    float* attn = ctx  + (size_t)BATCH * HEADS * SEQ * DIM;   // 4,194,304
    float* mask = attn + (size_t)BATCH * HEADS * SEQ * SEQ;   // +134,217,728

    const int blocks = BATCH * HEADS * (SEQ / 16);            // 4096
    fused_topk_attn<<<blocks, 256, 0, stream>>>(q, k, v, ctx, attn, mask);
}